// Cace_74569222193773
// MI455X (gfx1250) — compile-verified
//
#include <hip/hip_runtime.h>

// ---------------- problem constants ----------------
#define KNODES 2000
#define KEDGES 50000
#define KNL    20
#define KRBF   8
#define KRB    8
#define KCC    9
#define KLC    180            // KNL * KCC
#define KSTRIDE 1440          // KLC * 8
#define KCUT   5.5f
#define KMPN   0.2f           // 1/sqrt(25)
#define KPI    3.14159265358979323846f

typedef float v2f __attribute__((ext_vector_type(2)));
typedef float v8f __attribute__((ext_vector_type(8)));

__constant__ int c_LX[KNL] = {0, 1,0,0, 2,1,1,0,0,0, 3,2,2,1,1,1,0,0,0,0};
__constant__ int c_LY[KNL] = {0, 0,1,0, 0,1,0,2,1,0, 0,1,0,2,1,0,3,2,1,0};
__constant__ int c_LZ[KNL] = {0, 0,0,1, 0,0,1,0,1,2, 0,0,1,0,1,2,0,1,2,3};

__device__ __forceinline__ float poly_cut(float r) {
  float u = r / KCUT;
  float u2 = u * u, u4 = u2 * u2, u6 = u4 * u2;
  float f = 1.f - 28.f * u6 + 48.f * u6 * u - 21.f * u4 * u4;
  return (u < 1.f) ? f : 0.f;
}

// ============================================================
// Pad 8x8 weight matrices to 8x16 (cols 8..15 = 0) so WMMA B-operand
// loads need no divergence.  Layout in P (floats):
//   [   0.. 511] pad(Wrad)
//   [ 512..1023] pad(Wrad) * MP_NORM
//   [1024..1535] pad(Wmem)
// index within a 512-block: d*128 + r*16 + col
// ============================================================
__global__ void __launch_bounds__(256) pad_weights(
    const float* __restrict__ Wrad, const float* __restrict__ Wmem,
    float* __restrict__ P) {
  int t = blockIdx.x * blockDim.x + threadIdx.x;
  if (t >= 1536) return;
  int m = t >> 9;
  int i = t & 511;
  int d = i >> 7;
  int r = (i >> 4) & 7;
  int col = i & 15;
  float v = 0.f;
  if (col < 8) {
    float w = (m == 2) ? Wmem[d * 64 + r * 8 + col] : Wrad[d * 64 + r * 8 + col];
    v = (m == 1) ? w * KMPN : w;
  }
  P[t] = v;
}

// ============================================================
// Edge pass 1: scatter edge_attri into A0raw [n][lc][r]
// ============================================================
__global__ void __launch_bounds__(192) edge_pass1(
    const float* __restrict__ pos, const float* __restrict__ shifts,
    const float* __restrict__ Wemb, const int* __restrict__ species,
    const int* __restrict__ eidx, float* __restrict__ A0) {
  __shared__ float s_rad[KRBF], s_ang[KNL], s_enc[KCC];
  __shared__ float s_len, s_fcut, s_ux, s_uy, s_uz;
  int e = blockIdx.x;
  int t = threadIdx.x;
  int snd = eidx[e];
  int rcv = eidx[KEDGES + e];
  if (t == 0) {
    float vx = pos[rcv * 3 + 0] - pos[snd * 3 + 0] + shifts[e * 3 + 0];
    float vy = pos[rcv * 3 + 1] - pos[snd * 3 + 1] + shifts[e * 3 + 1];
    float vz = pos[rcv * 3 + 2] - pos[snd * 3 + 2] + shifts[e * 3 + 2];
    float len = sqrtf(vx * vx + vy * vy + vz * vz);
    float inv = 1.f / (len + 1e-9f);
    s_ux = vx * inv; s_uy = vy * inv; s_uz = vz * inv;
    s_len = len;
    s_fcut = poly_cut(len);
  }
  __syncthreads();
  if (t < KRBF) {
    float len = s_len;
    s_rad[t] = sqrtf(2.f / KCUT) * sinf((float)(t + 1) * KPI * len / KCUT) / len;
  } else if (t >= 32 && t < 32 + KNL) {
    int l = t - 32;
    float ux = s_ux, uy = s_uy, uz = s_uz;
    float p = 1.f;
    for (int i = 0; i < c_LX[l]; ++i) p *= ux;
    for (int i = 0; i < c_LY[l]; ++i) p *= uy;
    for (int i = 0; i < c_LZ[l]; ++i) p *= uz;
    s_ang[l] = p;
  } else if (t >= 64 && t < 64 + KCC) {
    int c = t - 64;
    s_enc[c] = Wemb[species[snd] * 3 + c / 3] * Wemb[species[rcv] * 3 + c % 3];
  }
  __syncthreads();
  if (t < KLC) {
    int l = t / 9, c = t - l * 9;
    float v = s_ang[l] * s_enc[c] * s_fcut;
    float* p = A0 + (size_t)rcv * KSTRIDE + t * 8;
#pragma unroll
    for (int r = 0; r < KRBF; ++r) atomicAdd(p + r, v * s_rad[r]);
  }
}

// ============================================================
// Edge pass 2: scatter msg_bchi -> Abchi [n][lc][r]
//              scatter msg_ar   -> Aar   [n][lc][b]
// ============================================================
__global__ void __launch_bounds__(192) edge_pass2(
    const float* __restrict__ pos, const float* __restrict__ shifts,
    const float* __restrict__ Wemb, const int* __restrict__ species,
    const int* __restrict__ eidx, const float* __restrict__ Amat,
    const float* __restrict__ chi, const float* __restrict__ War,
    float* __restrict__ Abchi, float* __restrict__ Aar) {
  __shared__ float s_rad[KRBF], s_ang[KNL], s_enc[KCC], s_chi[KCC], s_fr[KRB];
  __shared__ float s_len, s_fcut, s_ux, s_uy, s_uz;
  int e = blockIdx.x;
  int t = threadIdx.x;
  int snd = eidx[e];
  int rcv = eidx[KEDGES + e];
  // warm the caches for the A[send] gather while preliminaries compute
  if (t < KLC) __builtin_prefetch(Amat + (size_t)snd * KSTRIDE + t * 8, 0, 0);
  if (t == 0) {
    float vx = pos[rcv * 3 + 0] - pos[snd * 3 + 0] + shifts[e * 3 + 0];
    float vy = pos[rcv * 3 + 1] - pos[snd * 3 + 1] + shifts[e * 3 + 1];
    float vz = pos[rcv * 3 + 2] - pos[snd * 3 + 2] + shifts[e * 3 + 2];
    float len = sqrtf(vx * vx + vy * vy + vz * vz);
    float inv = 1.f / (len + 1e-9f);
    s_ux = vx * inv; s_uy = vy * inv; s_uz = vz * inv;
    s_len = len;
    s_fcut = poly_cut(len);
  }
  __syncthreads();
  if (t < KRBF) {
    float len = s_len;
    s_rad[t] = sqrtf(2.f / KCUT) * sinf((float)(t + 1) * KPI * len / KCUT) / len;
  } else if (t >= 32 && t < 32 + KNL) {
    int l = t - 32;
    float ux = s_ux, uy = s_uy, uz = s_uz;
    float p = 1.f;
    for (int i = 0; i < c_LX[l]; ++i) p *= ux;
    for (int i = 0; i < c_LY[l]; ++i) p *= uy;
    for (int i = 0; i < c_LZ[l]; ++i) p *= uz;
    s_ang[l] = p;
  } else if (t >= 64 && t < 64 + KCC) {
    int c = t - 64;
    s_enc[c] = Wemb[species[snd] * 3 + c / 3] * Wemb[species[rcv] * 3 + c % 3];
  } else if (t >= 96 && t < 96 + KCC) {
    s_chi[t - 96] = chi[snd * KCC + (t - 96)];
  }
  __syncthreads();
  if (t < KRB) {  // fr[b] = fcut * sum_r rad[r] * W_ar[r][b]
    float acc = 0.f;
#pragma unroll
    for (int r = 0; r < KRBF; ++r) acc += s_rad[r] * War[r * KRB + t];
    s_fr[t] = acc * s_fcut;
  }
  __syncthreads();
  if (t < KLC) {
    int l = t / 9, c = t - l * 9;
    size_t base = (size_t)rcv * KSTRIDE + t * 8;
    float v = s_ang[l] * s_enc[c] * s_fcut * s_chi[c];
#pragma unroll
    for (int r = 0; r < KRBF; ++r) atomicAdd(Abchi + base + r, v * s_rad[r]);
    const float* As = Amat + (size_t)snd * KSTRIDE + t * 8;
#pragma unroll
    for (int b = 0; b < KRB; ++b) atomicAdd(Aar + base + b, As[b] * s_fr[b]);
  }
}

// ============================================================
// WMMA GEMM pass 1 (per degree D, compile-time tile geometry):
//   A[n][lc][b] = sum_r A0raw[n][lc][r] * Wrad[D][r][b]
// Rows = (n, lc-within-degree); per-degree row counts are exact
// multiples of 16 => no padding, no bounds checks.  The 8 D-rows per
// lane are consecutive, and CNT >= 9 > 8, so at most one node-boundary
// wrap occurs across the unrolled store loop: one division total.
// ============================================================
template <int D, int CNT, int LOFF, int NT>
__global__ void __launch_bounds__(128) gemm_rt_t(const float* __restrict__ X,
                                                 const float* __restrict__ P,
                                                 float* __restrict__ Aout) {
  int wave = blockIdx.x * 4 + (threadIdx.x >> 5);
  if (wave >= NT) return;
  int lane = threadIdx.x & 31;
  int row0 = wave * 16;
  int half = lane >> 4, l16 = lane & 15;
  int kb = half * 2;
  int rowA = row0 + l16;
  int n = rowA / CNT;
  int i = rowA - n * CNT;
  const float* Xr = X + n * KSTRIDE + (LOFF + i) * 8;
  const float* W = P + D * 128;  // padded 8x16
  v8f c = {0.f, 0.f, 0.f, 0.f, 0.f, 0.f, 0.f, 0.f};
#pragma unroll
  for (int kk = 0; kk < 8; kk += 4) {
    v2f a = *(const v2f*)(Xr + kk + kb);
    v2f b;
    b.x = W[(kk + kb) * 16 + l16];
    b.y = W[(kk + kb + 1) * 16 + l16];
    c = __builtin_amdgcn_wmma_f32_16x16x4_f32(false, a, false, b, (short)0, c,
                                              false, false);
  }
  if (l16 < 8) {
    int rbase = row0 + half * 8;
    int n0 = rbase / CNT;
    int j0 = rbase - n0 * CNT;
    int base0 = n0 * KSTRIDE + (LOFF + j0) * 8 + l16;
#pragma unroll
    for (int ii = 0; ii < 8; ++ii) {
      int tt = j0 + ii;
      bool w = tt >= CNT;
      int addr = base0 + ii * 8 + (w ? (KSTRIDE - CNT * 8) : 0);
      Aout[addr] = c[ii];
    }
  }
}

// ============================================================
// WMMA GEMM pass 2 (message-passing update), K = 16 via 4 chained WMMAs:
//   A2 = Aar*MPN + RT(Abchi)*MPN + MEM(A)
// C operand pre-loaded with Aar*MPN (lanes col>=8 load a duplicate
// column; their outputs are never stored).  Written in place over Aar.
// ============================================================
template <int D, int CNT, int LOFF, int NT>
__global__ void __launch_bounds__(128) gemm_mp_t(const float* __restrict__ Xb,
                                                 const float* __restrict__ Amat,
                                                 const float* __restrict__ P,
                                                 float* __restrict__ Aar) {
  int wave = blockIdx.x * 4 + (threadIdx.x >> 5);
  if (wave >= NT) return;
  int lane = threadIdx.x & 31;
  int row0 = wave * 16;
  int half = lane >> 4, l16 = lane & 15;
  int kb = half * 2;
  int col = l16 & 7;
  // D/C rows for this lane: rbase..rbase+7 (consecutive, <=1 node wrap)
  int rbase = row0 + half * 8;
  int n0 = rbase / CNT;
  int j0 = rbase - n0 * CNT;
  int cbase = n0 * KSTRIDE + (LOFF + j0) * 8;
  v8f c;
#pragma unroll
  for (int ii = 0; ii < 8; ++ii) {
    int tt = j0 + ii;
    bool w = tt >= CNT;
    int addr = cbase + ii * 8 + (w ? (KSTRIDE - CNT * 8) : 0);
    c[ii] = Aar[addr + col] * KMPN;
  }
  int rowA = row0 + l16;
  int n = rowA / CNT;
  int i = rowA - n * CNT;
  int xoff = n * KSTRIDE + (LOFF + i) * 8;
  const float* W1 = P + 512 + D * 128;   // pad(Wrad)*MPN
  const float* W2 = P + 1024 + D * 128;  // pad(Wmem)
#pragma unroll
  for (int ch = 0; ch < 4; ++ch) {
    int kk = (ch & 1) * 4;
    const float* Xs = (ch < 2) ? Xb : Amat;
    const float* W = (ch < 2) ? W1 : W2;
    v2f a = *(const v2f*)(Xs + xoff + kk + kb);
    v2f b;
    b.x = W[(kk + kb) * 16 + l16];
    b.y = W[(kk + kb + 1) * 16 + l16];
    c = __builtin_amdgcn_wmma_f32_16x16x4_f32(false, a, false, b, (short)0, c,
                                              false, false);
  }
  if (l16 < 8) {
#pragma unroll
    for (int ii = 0; ii < 8; ++ii) {
      int tt = j0 + ii;
      bool w = tt >= CNT;
      int addr = cbase + ii * 8 + (w ? (KSTRIDE - CNT * 8) : 0);
      Aar[addr + l16] = c[ii];
    }
  }
}

// ============================================================
// Symmetrize: B[n][b][k][c] -> out[...,tsel]; optional chi[n][c]
// one thread per (n, c, b)
// ============================================================
__global__ void __launch_bounds__(256) symm_kernel(
    const float* __restrict__ Amat, const float* __restrict__ Wchi,
    float* __restrict__ out, float* __restrict__ chi, int tsel, int do_chi) {
  constexpr int DEG[KNL] = {0, 1, 1, 1, 2, 2, 2, 2, 2, 2,
                            3, 3, 3, 3, 3, 3, 3, 3, 3, 3};
  constexpr float MULTI[KNL] = {1.f, 1.f, 1.f, 1.f, 1.f, 2.f, 2.f,
                                1.f, 2.f, 1.f, 1.f, 3.f, 3.f, 3.f,
                                6.f, 3.f, 1.f, 3.f, 3.f, 1.f};
  int idx = blockIdx.x * blockDim.x + threadIdx.x;
  if (idx >= KNODES * 72) return;
  int n = idx / 72;
  int rem = idx - n * 72;
  int cc = rem >> 3;
  int b = rem & 7;
  const float* Ap = Amat + (size_t)n * KSTRIDE + cc * 8 + b;  // l stride = 72
  float accs[4] = {0.f, 0.f, 0.f, 0.f};
  float b1 = Ap[0];
#pragma unroll
  for (int l = 0; l < KNL; ++l) {
    float v = Ap[l * 72];
    accs[DEG[l]] += v * v * MULTI[l];
  }
  float Bv[5] = {b1, accs[0], accs[1], accs[2], accs[3]};
#pragma unroll
  for (int k = 0; k < 5; ++k)
    out[(((size_t)(n * 8 + b) * 5 + k) * 9 + cc) * 2 + tsel] = Bv[k];
  if (do_chi) {
    float p = 0.f;
#pragma unroll
    for (int k = 0; k < 5; ++k) p += Bv[k] * Wchi[b * 5 + k];
    atomicAdd(&chi[n * KCC + cc], p);
  }
}

// ============================================================
extern "C" void kernel_launch(void* const* d_in, const int* in_sizes, int n_in,
                              void* d_out, int out_size, void* d_ws,
                              size_t ws_size, hipStream_t stream) {
  const float* pos = (const float*)d_in[0];
  const float* shifts = (const float*)d_in[1];
  const float* Wemb = (const float*)d_in[2];
  const float* Wrad = (const float*)d_in[3];
  const float* Wmem = (const float*)d_in[4];
  const float* War = (const float*)d_in[5];
  const float* Wchi = (const float*)d_in[6];
  const int* species = (const int*)d_in[7];
  const int* eidx = (const int*)d_in[8];
  float* out = (float*)d_out;
  float* ws = (float*)d_ws;

  const size_t szA = (size_t)KNODES * KSTRIDE;  // 2,880,000 floats
  float* A0 = ws;             // edge accumulator (pass1: A0raw, pass2: Abchi)
  float* Amat = ws + szA;     // A after radial transform
  float* Aar = ws + 2 * szA;  // A_ar accumulator, overwritten in place with A2
  float* chi = ws + 3 * szA;  // 18000 floats
  float* P = chi + (size_t)KNODES * KCC;  // 1536 floats padded weights

  hipMemsetAsync(A0, 0, szA * sizeof(float), stream);
  hipMemsetAsync(Aar, 0, szA * sizeof(float), stream);
  hipMemsetAsync(chi, 0, (size_t)KNODES * KCC * sizeof(float), stream);
  pad_weights<<<6, 256, 0, stream>>>(Wrad, Wmem, P);

  // ---- t = 0 ----
  edge_pass1<<<KEDGES, 192, 0, stream>>>(pos, shifts, Wemb, species, eidx, A0);
  gemm_rt_t<0, 9, 0, 1125><<<282, 128, 0, stream>>>(A0, P, Amat);
  gemm_rt_t<1, 27, 9, 3375><<<844, 128, 0, stream>>>(A0, P, Amat);
  gemm_rt_t<2, 54, 36, 6750><<<1688, 128, 0, stream>>>(A0, P, Amat);
  gemm_rt_t<3, 90, 90, 11250><<<2813, 128, 0, stream>>>(A0, P, Amat);
  symm_kernel<<<(KNODES * 72 + 255) / 256, 256, 0, stream>>>(Amat, Wchi, out,
                                                             chi, 0, 1);
  // ---- t = 1 (message passing round) ----
  hipMemsetAsync(A0, 0, szA * sizeof(float), stream);
  edge_pass2<<<KEDGES, 192, 0, stream>>>(pos, shifts, Wemb, species, eidx,
                                         Amat, chi, War, A0, Aar);
  gemm_mp_t<0, 9, 0, 1125><<<282, 128, 0, stream>>>(A0, Amat, P, Aar);
  gemm_mp_t<1, 27, 9, 3375><<<844, 128, 0, stream>>>(A0, Amat, P, Aar);
  gemm_mp_t<2, 54, 36, 6750><<<1688, 128, 0, stream>>>(A0, Amat, P, Aar);
  gemm_mp_t<3, 90, 90, 11250><<<2813, 128, 0, stream>>>(A0, Amat, P, Aar);
  symm_kernel<<<(KNODES * 72 + 255) / 256, 256, 0, stream>>>(Aar, Wchi, out,
                                                             chi, 1, 0);
}